// GCNLayer_73065983640003
// MI455X (gfx1250) — compile-verified
//
#include <hip/hip_runtime.h>
#include <hip/hip_bf16.h>
#include <stdint.h>

// ---------------------------------------------------------------------------
// GCN 2-relation aggregation for MI455X (gfx1250).
//   out[dst] += feat[src] * norm   (per edge, two edge sets, summed)
//
// Memory-bound op (AI ~ 0.25 FLOP/B): whole working set (~65 MB) fits in the
// 192 MB L2, so the strategy is L2-resident gather + native f32 L2 atomics
// for the scatter. Wave32-per-edge mapping: lane l covers features
// {l, l+32, l+64} -> every vector load / atomic touches a contiguous 128B
// segment (perfect coalescing). Edge metadata (src/dst/norm) is staged into
// LDS via the Tensor Data Mover (tensor_load_to_lds + s_wait_tensorcnt),
// one 1-D descriptor per array per 256-edge block.
// ---------------------------------------------------------------------------

#define D_FEAT 96
#define TILE   256   // edges per block (800000 % 256 == 0, but guarded anyway)

typedef unsigned int u32x4 __attribute__((ext_vector_type(4)));
typedef int          i32x8 __attribute__((ext_vector_type(8)));
typedef int          i32x4 __attribute__((ext_vector_type(4)));

#if __has_builtin(__builtin_amdgcn_tensor_load_to_lds)
#define HAVE_TDM 1
#else
#define HAVE_TDM 0
#endif

__device__ __forceinline__ void atomic_add_f32(float* p, float v) {
  // Relaxed, device-scope f32 add -> global_atomic_add_f32
  // (no-return form, tracked by STOREcnt; no wait needed in the loop).
  __hip_atomic_fetch_add(p, v, __ATOMIC_RELAXED, __HIP_MEMORY_SCOPE_AGENT);
}

#if HAVE_TDM
// Issue one TDM descriptor copying `nelem` 4-byte elements from gptr into LDS
// at byte offset lds_off. 1-D tile: tile_dim0 = tensor_dim0 = nelem.
// This toolchain's builtin takes 6 args:
//   (u32x4 g0, i32x8 g1, i32x4 g2, i32x4 g3, i32x8 extra, i32 cpol)
__device__ __forceinline__ void tdm_load_1d(uint32_t lds_off,
                                            const void* gptr,
                                            uint32_t nelem) {
  uint64_t ga = (uint64_t)(uintptr_t)gptr;

  u32x4 g0;
  g0.x = 1u;                                               // count=1, user mode
  g0.y = lds_off;                                          // lds_addr (bytes)
  g0.z = (uint32_t)ga;                                     // global_addr[31:0]
  g0.w = ((uint32_t)(ga >> 32) & 0x1FFFFFFu) | (2u << 30); // addr[56:32], type=2

  i32x8 g1;
  g1[0] = (int)(2u << 16);                    // data_size=2 -> 4 bytes; no mask
  g1[1] = (int)((nelem & 0xFFFFu) << 16);     // tensor_dim0[15:0] @ bits 63:48
  g1[2] = (int)((nelem >> 16) | (1u << 16));  // tensor_dim0[31:16] | dim1 lo = 1
  g1[3] = (int)((nelem & 0xFFFFu) << 16);     // dim1 hi = 0 | tile_dim0 = nelem
  g1[4] = 1;                                  // tile_dim1 = 1, tile_dim2 = 0
  g1[5] = (int)nelem;                         // tensor_dim0_stride lo
  g1[6] = 0;                                  // stride hi | dim1_stride lo
  g1[7] = 0;                                  // dim1_stride hi

  i32x4 g2 = {1, 1, 0, 0};                    // tensor_dim2=1, tensor_dim3=1
  i32x4 g3 = {0, (int)(1u << 16), 0, 0};      // tensor_dim4=1, tile_dim4=0

  i32x8 gx = {0, 0, 0, 0, 0, 0, 0, 0};        // extra descriptor words (unused)

  __builtin_amdgcn_tensor_load_to_lds(g0, g1, g2, g3, gx, /*cpol=*/0);
}
#endif

// ---------------------------------------------------------------------------
// One 256-edge tile per block; 8 waves; one wave32 per edge.
// (Defined first so the disassembly snippet shows this kernel.)
__global__ void __launch_bounds__(TILE)
gcn_edge_kernel(const float* __restrict__ feat,
                const float* __restrict__ norm,
                const int*   __restrict__ src,
                const int*   __restrict__ dst,
                float*       __restrict__ out,
                int n_edges) {
  __shared__ int   s_src[TILE];
  __shared__ int   s_dst[TILE];
  __shared__ float s_norm[TILE];

  const int base = blockIdx.x * TILE;
  const int nloc = min(TILE, n_edges - base);
  const int tid  = threadIdx.x;

#if HAVE_TDM
  if (tid == 0) {
    tdm_load_1d((uint32_t)(uintptr_t)s_src,  src  + base, (uint32_t)nloc);
    tdm_load_1d((uint32_t)(uintptr_t)s_dst,  dst  + base, (uint32_t)nloc);
    tdm_load_1d((uint32_t)(uintptr_t)s_norm, norm + base, (uint32_t)nloc);
    __builtin_amdgcn_s_wait_tensorcnt(0);   // TDM writes LDS -> TENSORcnt
  }
  __syncthreads();
#else
  if (tid < nloc) {
    s_src[tid]  = src[base + tid];
    s_dst[tid]  = dst[base + tid];
    s_norm[tid] = norm[base + tid];
  }
  __syncthreads();
#endif

  const int wave  = tid >> 5;            // 0..7 (wave32)
  const int lane  = tid & 31;
  const int e0    = wave * (TILE / 8);   // 32 edges per wave
  const int e1    = min(e0 + (TILE / 8), nloc);

#pragma unroll 4
  for (int e = e0; e < e1; ++e) {
    const int   s = s_src[e];            // LDS broadcast reads (uniform addr)
    const int   d = s_dst[e];
    const float w = s_norm[e];

    const float* frow = feat + (size_t)s * D_FEAT;
    float*       orow = out  + (size_t)d * D_FEAT;

    // 3 coalesced b32 gathers (contiguous 128B per wave per load; L2-resident)
    float f0 = frow[lane];
    float f1 = frow[lane + 32];
    float f2 = frow[lane + 64];

    // 3 coalesced fire-and-forget f32 atomics (resolve in L2)
    atomic_add_f32(orow + lane,      f0 * w);
    atomic_add_f32(orow + lane + 32, f1 * w);
    atomic_add_f32(orow + lane + 64, f2 * w);
  }
}

// ---------------------------------------------------------------------------
__global__ void gcn_zero_kernel(float4* __restrict__ out, int n4) {
  int i = blockIdx.x * blockDim.x + threadIdx.x;
  int stride = gridDim.x * blockDim.x;
  float4 z = make_float4(0.f, 0.f, 0.f, 0.f);
  for (; i < n4; i += stride) out[i] = z;
}

// ---------------------------------------------------------------------------
extern "C" void kernel_launch(void* const* d_in, const int* in_sizes, int n_in,
                              void* d_out, int out_size, void* d_ws, size_t ws_size,
                              hipStream_t stream) {
  const float* feat  = (const float*)d_in[0];
  const float* norm0 = (const float*)d_in[1];
  const float* norm1 = (const float*)d_in[2];
  const int*   src0  = (const int*)d_in[3];
  const int*   dst0  = (const int*)d_in[4];
  const int*   src1  = (const int*)d_in[5];
  const int*   dst1  = (const int*)d_in[6];
  float* out = (float*)d_out;

  const int n_edges0 = in_sizes[3];
  const int n_edges1 = in_sizes[5];

  // Zero-init output (harness poisons d_out; atomics accumulate into it).
  const int n4 = out_size / 4;            // 50000*96 / 4 float4's
  gcn_zero_kernel<<<1024, 256, 0, stream>>>((float4*)out, n4);

  const int g0 = (n_edges0 + TILE - 1) / TILE;
  gcn_edge_kernel<<<g0, TILE, 0, stream>>>(feat, norm0, src0, dst0, out, n_edges0);

  const int g1 = (n_edges1 + TILE - 1) / TILE;
  gcn_edge_kernel<<<g1, TILE, 0, stream>>>(feat, norm1, src1, dst1, out, n_edges1);
}